// LearnBFARPolicy_59871844106714
// MI455X (gfx1250) — compile-verified
//
#include <hip/hip_runtime.h>

#define RESV   0.0596f
#define TRIMV  5.0f
#define HUBERV 1.0f
#define NB 8
#define NA 400
#define NR 1024
#define NM 4096
#define NP 1600           // NA * KPEAK
#define NITER 50
#define SPLIT 16          // blocks per batch in the ICP step kernel
#define NTILE (NP / 16)   // 100 point tiles

typedef __attribute__((ext_vector_type(2))) float v2f;
typedef __attribute__((ext_vector_type(8))) float v8f;

// ---------------------------------------------------------------------------
// insert (v,ix) into a descending-sorted top-4 list
// ---------------------------------------------------------------------------
__device__ __forceinline__ void ins4(float v, int ix, float tv[4], int ti[4]) {
  if (v <= tv[3]) return;
  if (v > tv[1]) {
    if (v > tv[0]) {
      tv[3]=tv[2]; ti[3]=ti[2]; tv[2]=tv[1]; ti[2]=ti[1];
      tv[1]=tv[0]; ti[1]=ti[0]; tv[0]=v;     ti[0]=ix;
    } else {
      tv[3]=tv[2]; ti[3]=ti[2]; tv[2]=tv[1]; ti[2]=ti[1];
      tv[1]=v;     ti[1]=ix;
    }
  } else {
    if (v > tv[2]) { tv[3]=tv[2]; ti[3]=ti[2]; tv[2]=v; ti[2]=ix; }
    else           { tv[3]=v;     ti[3]=ix; }
  }
}

// ---------------------------------------------------------------------------
// Stage 1: BFAR extraction. One wave32 per (b, azimuth) row of 1024 bins.
// top-k of score == top-k of raw fft (threshold is row-constant).
// ---------------------------------------------------------------------------
__global__ void __launch_bounds__(256) bfar_extract(
    const float* __restrict__ fft, const float* __restrict__ az,
    const float* __restrict__ params,
    float* __restrict__ scan, float* __restrict__ wout) {
  const int row  = (blockIdx.x * blockDim.x + threadIdx.x) >> 5;  // 0..3199
  const int lane = threadIdx.x & 31;
  const int b    = row / NA;
  const int arow = row - b * NA;

  float tv[4] = {-3.4e38f, -3.4e38f, -3.4e38f, -3.4e38f};
  int   ti[4] = {0, 0, 0, 0};
  float sum = 0.f;
  const float* rp = fft + (size_t)row * NR;
  #pragma unroll
  for (int k = 0; k < NR / 32; ++k) {
    int ix = k * 32 + lane;
    float v = rp[ix];
    sum += v;
    ins4(v, ix, tv, ti);
  }
  #pragma unroll
  for (int off = 1; off < 32; off <<= 1) {
    sum += __shfl_xor(sum, off, 32);
    float ov[4]; int oi[4];
    #pragma unroll
    for (int k = 0; k < 4; ++k) {
      ov[k] = __shfl_xor(tv[k], off, 32);
      oi[k] = __shfl_xor(ti[k], off, 32);
    }
    #pragma unroll
    for (int k = 0; k < 4; ++k) ins4(ov[k], oi[k], tv, ti);
  }
  if (lane == 0) {
    float pa  = fmaxf(params[0], 0.f);
    float pb  = fmaxf(params[1], 0.f);
    float thr = pa * (sum * (1.0f / NR)) + pb;
    float azv = az[b * NA + arow];
    float c = cosf(azv), s = sinf(azv);
    int base = b * NP + arow * 4;
    #pragma unroll
    for (int k = 0; k < 4; ++k) {
      float rng = ((float)ti[k] + 0.5f) * RESV;
      scan[2 * (base + k) + 0] = rng * c;
      scan[2 * (base + k) + 1] = rng * s;
      wout[base + k] = fmaxf(tv[k] - thr, 0.f);
    }
  }
}

// ---------------------------------------------------------------------------
// Stage 2: normalize 2D map normals (components 3:5 of map_pc, stride 6)
// ---------------------------------------------------------------------------
__global__ void norm_kernel(const float* __restrict__ map_pc,
                            float* __restrict__ nrm) {
  int i = blockIdx.x * blockDim.x + threadIdx.x;
  if (i >= NB * NM) return;
  const float* p = map_pc + (size_t)i * 6;
  float nx = p[3], ny = p[4];
  float inv = 1.0f / fmaxf(sqrtf(nx * nx + ny * ny), 1e-12f);
  nrm[2 * i + 0] = nx * inv;
  nrm[2 * i + 1] = ny * inv;
}

// ---------------------------------------------------------------------------
// init: state[b] = (theta0, tx0, ty0) from T_init
// ---------------------------------------------------------------------------
__global__ void icp_init(const float* __restrict__ Tinit,
                         float* __restrict__ state) {
  int b = threadIdx.x;
  if (b < NB) {
    const float* T = Tinit + b * 16;
    state[b * 3 + 0] = atan2f(T[4], T[0]);  // T[1][0], T[0][0]
    state[b * 3 + 1] = T[3];                // T[0][3]
    state[b * 3 + 2] = T[7];                // T[1][3]
  }
}

// ---------------------------------------------------------------------------
// One ICP iteration, NN + Gauss-Newton partials. Grid = NB*SPLIT blocks of
// 256; each wave owns one 16-point tile. WMMA operand layout (f32 16x16x4):
//   A (16x4):  lanes 0-15 hold row m=lane: a.x=K0=px, a.y=K1=py;
//              lanes 16-31 hold row m=lane-16: a.x=K2=-0.5, a.y=K3=0.
//   B (4x16):  bv.x: lanes 0-15 = K0 row (qx[n]), lanes 16-31 = K2 row (|q|^2)
//              bv.y: lanes 0-15 = K1 row (qy[n]), lanes 16-31 = K3 row (x0 pad)
//   D = p.q - 0.5*|q|^2  ->  nearest neighbour = argmax over columns.
//   D (16x16): VGPR r: lanes 0-15 -> row r, lanes 16-31 -> row r+8, col=lane&15
// ---------------------------------------------------------------------------
__global__ void __launch_bounds__(256) icp_step(
    const float* __restrict__ scan, const float* __restrict__ wbuf,
    const float* __restrict__ map_pc, const float* __restrict__ nrm,
    const float* __restrict__ state, float* __restrict__ part_out) {
  __shared__ float s_qx[NM];
  __shared__ float s_qy[NM];
  __shared__ float s_q2[NM];
  __shared__ int   s_nnw[8][16];   // per-wave NN scratch
  __shared__ float s_part[8][9];   // per-wave partial sums

  const int b    = blockIdx.x / SPLIT;
  const int part = blockIdx.x - b * SPLIT;
  const int tid  = threadIdx.x;
  const int lane = tid & 31;
  const int wid  = tid >> 5;
  const int half = lane >> 4;
  const int col  = lane & 15;

  for (int i = tid; i < NM; i += 256) {
    const float* p = map_pc + ((size_t)b * NM + i) * 6;
    float x = p[0], y = p[1];
    s_qx[i] = x; s_qy[i] = y; s_q2[i] = x * x + y * y;
  }
  __syncthreads();

  const float th = state[b * 3 + 0];
  const float tx = state[b * 3 + 1];
  const float ty = state[b * 3 + 2];
  const float c = cosf(th), s = sinf(th);

  const float* sc = scan + (size_t)b * NP * 2;
  const float* wb = wbuf + (size_t)b * NP;
  const float* nb = nrm + (size_t)b * NM * 2;

  float aA[9] = {0, 0, 0, 0, 0, 0, 0, 0, 0};

  const int pt = part * 8 + wid;     // wave-slot: 0..127, tiles 0..99 valid
  if (pt < NTILE) {
    // ---- Phase A: WMMA argmax NN (EXEC all-1s; wave-uniform branch) ----
    const int pi = pt * 16 + col;
    const float sx = sc[2 * pi], sy = sc[2 * pi + 1];
    const float px = c * sx - s * sy + tx;
    const float py = s * sx + c * sy + ty;
    v2f a;
    a.x = half ? -0.5f : px;   // K0 | K2
    a.y = half ?  0.0f : py;   // K1 | K3
    const float* bx_base = half ? s_q2 : s_qx;  // hoisted per-lane base
    float bk[8]; int bi[8];
    #pragma unroll
    for (int r = 0; r < 8; ++r) { bk[r] = -3.4e38f; bi[r] = 0; }
    #pragma unroll 2
    for (int t = 0; t < NM / 16; ++t) {
      int o = t * 16 + col;
      v2f bv;
      bv.x = bx_base[o];       // K0 row = qx | K2 row = |q|^2
      bv.y = s_qy[o];          // K1 row = qy | K3 row: multiplied by A.K3==0
      v8f cz = {0.f, 0.f, 0.f, 0.f, 0.f, 0.f, 0.f, 0.f};
      v8f d = __builtin_amdgcn_wmma_f32_16x16x4_f32(
          false, a, false, bv, (short)0, cz, false, false);
      #pragma unroll
      for (int r = 0; r < 8; ++r) {
        bool tk = d[r] > bk[r];
        bk[r] = tk ? d[r] : bk[r];
        bi[r] = tk ? o : bi[r];
      }
    }
    // branchless argmax reduce across the 16-lane half (tie -> lower index)
    #pragma unroll
    for (int off = 1; off < 16; off <<= 1) {
      #pragma unroll
      for (int r = 0; r < 8; ++r) {
        float ok = __shfl_xor(bk[r], off, 32);
        int   oi = __shfl_xor(bi[r], off, 32);
        bool tk = (ok > bk[r]) | ((ok == bk[r]) & (oi < bi[r]));
        bk[r] = tk ? ok : bk[r];
        bi[r] = tk ? oi : bi[r];
      }
    }
    if (lane == 0) {
      #pragma unroll
      for (int r = 0; r < 8; ++r) s_nnw[wid][r] = bi[r];        // rows 0-7
    } else if (lane == 16) {
      #pragma unroll
      for (int r = 0; r < 8; ++r) s_nnw[wid][8 + r] = bi[r];    // rows 8-15
    }
    // ---- Phase B: lanes 0-15 own one point each (same-wave DS ordering) ----
    if (!half) {
      int j = s_nnw[wid][col];
      float ex = px - s_qx[j], ey = py - s_qy[j];
      float nx = nb[2 * j], ny = nb[2 * j + 1];
      float res = nx * ex + ny * ey;
      float d2 = ex * ex + ey * ey;
      float hub = fminf(1.0f, HUBERV / fmaxf(fabsf(res), 1e-12f));
      float wt = wb[pi] * hub * ((d2 < TRIMV * TRIMV) ? 1.f : 0.f);
      float drx = -s * sx - c * sy;
      float dry =  c * sx - s * sy;
      float j2 = nx * drx + ny * dry;
      aA[0] = wt * nx * nx;  aA[1] = wt * nx * ny;  aA[2] = wt * nx * j2;
      aA[3] = wt * ny * ny;  aA[4] = wt * ny * j2;  aA[5] = wt * j2 * j2;
      aA[6] = wt * nx * res; aA[7] = wt * ny * res; aA[8] = wt * j2 * res;
    }
  }
  // fixed-order wave reduction (no atomics -> bitwise deterministic)
  #pragma unroll
  for (int off = 1; off < 32; off <<= 1) {
    #pragma unroll
    for (int k = 0; k < 9; ++k) aA[k] += __shfl_xor(aA[k], off, 32);
  }
  if (lane == 0) {
    #pragma unroll
    for (int k = 0; k < 9; ++k) s_part[wid][k] = aA[k];
  }
  __syncthreads();
  if (tid < 9) {
    float v = 0.f;
    #pragma unroll
    for (int w = 0; w < 8; ++w) v += s_part[w][tid];
    part_out[(size_t)blockIdx.x * 9 + tid] = v;
  }
}

// ---------------------------------------------------------------------------
// Per-iteration 3x3 symmetric solve (fixed-order partial sum, Cramer)
// ---------------------------------------------------------------------------
__global__ void icp_solve(float* __restrict__ state,
                          const float* __restrict__ part) {
  int b = threadIdx.x;
  if (b >= NB) return;
  float m[9] = {0, 0, 0, 0, 0, 0, 0, 0, 0};
  #pragma unroll
  for (int p = 0; p < SPLIT; ++p) {
    #pragma unroll
    for (int k = 0; k < 9; ++k) m[k] += part[(size_t)(b * SPLIT + p) * 9 + k];
  }
  float a00 = m[0] + 1e-8f, a01 = m[1], a02 = m[2];
  float a11 = m[3] + 1e-8f, a12 = m[4], a22 = m[5] + 1e-8f;
  float g0 = m[6], g1 = m[7], g2 = m[8];
  float c00 = a11 * a22 - a12 * a12;
  float c01 = a02 * a12 - a01 * a22;
  float c02 = a01 * a12 - a02 * a11;
  float det = a00 * c00 + a01 * c01 + a02 * c02;
  float inv = 1.0f / det;
  float c11 = a00 * a22 - a02 * a02;
  float c12 = a01 * a02 - a00 * a12;
  float c22 = a00 * a11 - a01 * a01;
  float x0 = (c00 * g0 + c01 * g1 + c02 * g2) * inv;
  float x1 = (c01 * g0 + c11 * g1 + c12 * g2) * inv;
  float x2 = (c02 * g0 + c12 * g1 + c22 * g2) * inv;
  state[b * 3 + 0] -= x2;
  state[b * 3 + 1] -= x0;
  state[b * 3 + 2] -= x1;
}

// ---------------------------------------------------------------------------
// finalize: emit 4x4 transforms
// ---------------------------------------------------------------------------
__global__ void icp_final(const float* __restrict__ state,
                          float* __restrict__ out) {
  int b = threadIdx.x;
  if (b >= NB) return;
  float th = state[b * 3 + 0], tx = state[b * 3 + 1], ty = state[b * 3 + 2];
  float c = cosf(th), s = sinf(th);
  float* T = out + b * 16;
  T[0]  = c;   T[1]  = -s;  T[2]  = 0.f; T[3]  = tx;
  T[4]  = s;   T[5]  = c;   T[6]  = 0.f; T[7]  = ty;
  T[8]  = 0.f; T[9]  = 0.f; T[10] = 1.f; T[11] = 0.f;
  T[12] = 0.f; T[13] = 0.f; T[14] = 0.f; T[15] = 1.f;
}

// ---------------------------------------------------------------------------
// launch: all on `stream`, graph-capture safe, deterministic
// ---------------------------------------------------------------------------
extern "C" void kernel_launch(void* const* d_in, const int* in_sizes, int n_in,
                              void* d_out, int out_size, void* d_ws, size_t ws_size,
                              hipStream_t stream) {
  (void)in_sizes; (void)n_in; (void)out_size; (void)ws_size;
  const float* fft    = (const float*)d_in[0];  // [B, A, R]
  const float* az     = (const float*)d_in[1];  // [B, A]
  // d_in[2] az_timestamps: unused by reference
  const float* map_pc = (const float*)d_in[3];  // [B, M, 6]
  const float* Tinit  = (const float*)d_in[4];  // [B, 4, 4]
  const float* params = (const float*)d_in[5];  // [2]
  float* out = (float*)d_out;                   // [B, 4, 4]

  float* ws    = (float*)d_ws;
  float* scan  = ws;                               // B*NP*2
  float* wbuf  = scan + (size_t)NB * NP * 2;       // B*NP
  float* nrm   = wbuf + (size_t)NB * NP;           // B*NM*2
  float* state = nrm + (size_t)NB * NM * 2;        // B*3
  float* parts = state + (size_t)NB * 3;           // B*SPLIT*9

  bfar_extract<<<(NB * NA) / 8, 256, 0, stream>>>(fft, az, params, scan, wbuf);
  norm_kernel<<<(NB * NM + 255) / 256, 256, 0, stream>>>(map_pc, nrm);
  icp_init<<<1, 32, 0, stream>>>(Tinit, state);
  for (int it = 0; it < NITER; ++it) {
    icp_step<<<NB * SPLIT, 256, 0, stream>>>(scan, wbuf, map_pc, nrm, state, parts);
    icp_solve<<<1, 32, 0, stream>>>(state, parts);
  }
  icp_final<<<1, 32, 0, stream>>>(state, out);
}